// GATModel_70970039599393
// MI455X (gfx1250) — compile-verified
//
#include <hip/hip_runtime.h>
#include <hip/hip_bf16.h>

// ---------------------------------------------------------------------------
// GAT forward for MI455X (gfx1250, wave32).
//   N=50000 nodes, F=H*C=128 feats, H=4 heads, C=32, E=800000 edges (+N loops)
// Phases per layer:
//   1) xw = h @ W           -> fp32 WMMA 16x16x4, 64-row blocks, B-frag reuse x4
//   2) alpha_s/d per node   -> wave32 shuffle reduction (1 wave per head)
//   3) segment max over dst -> ordered-uint atomicMax (native u32 atomic)
//   4) exp + segment sum    -> atomicAdd f32
//   5) aggregation          -> 1 wave per edge, lane=channel, f32 atomics
//   6) bias + ELU
// Final: logits = h @ Wl + bl, log_softmax per node.
// All working buffers (~94 MB) fit in the 192 MB L2 -> scatter phase is
// L2-resident; HBM is touched roughly once.
// ---------------------------------------------------------------------------

typedef float v2f __attribute__((ext_vector_type(2)));
typedef float v8f __attribute__((ext_vector_type(8)));

#define FDIM 128
#define HEADS 4
#define CH 32
#define NCLS 40
#define NEG_SLOPE 0.2f

// rows per GEMM block (4 x 16-row WMMA tiles share each B fragment)
#define GROWS 64

// ------------------------- ordered-float encoding --------------------------
__device__ __forceinline__ unsigned fenc(float f) {
  unsigned u = __float_as_uint(f);
  return (u & 0x80000000u) ? ~u : (u | 0x80000000u);
}
__device__ __forceinline__ float fdec(unsigned k) {
  return (k & 0x80000000u) ? __uint_as_float(k & 0x7FFFFFFFu)
                           : __uint_as_float(~k);
}
// encoding of -inf (smallest possible key for any real value)
#define FENC_NEG_INF 0x007FFFFFu

// ------------------------------ GEMM (WMMA) --------------------------------
// out[N,128] = A[N,128] @ W[128,128].
// Block: 256 threads = 8 waves. Block owns a 64-row tile of A (staged in LDS
// with float4 loads; row stride 132 floats -> conflict-free column reads).
// Wave w owns output columns [16w,16w+16) and 4 row-tiles: each B fragment
// (2 VGPRs) is loaded once per k-step and reused by 4 WMMAs.
__global__ void gemm_xw_kernel(const float* __restrict__ A,
                               const float* __restrict__ W,
                               float* __restrict__ out, int N) {
  __shared__ float As[GROWS][132];
  const int row0 = blockIdx.x * GROWS;
  const int tid = threadIdx.x;

  // stage GROWSx128 A tile with float4 (GROWS*32 float4 / 256 threads)
  const float4 zero4 = make_float4(0.f, 0.f, 0.f, 0.f);
  for (int v = tid; v < GROWS * (FDIM / 4); v += 256) {
    int r = v >> 5;        // FDIM/4 == 32 float4 per row
    int c4 = v & 31;
    int gr = row0 + r;
    float4 val =
        (gr < N) ? ((const float4*)(A + (size_t)gr * FDIM))[c4] : zero4;
    *(float4*)(&As[r][c4 * 4]) = val;
  }
  __syncthreads();

  const int wave = tid >> 5;
  const int lane = tid & 31;
  const int col0 = wave * 16;
  const int half = lane >> 4;   // 0: K pair {k,k+1}; 1: K pair {k+2,k+3}
  const int l16 = lane & 15;

  v8f acc[4] = {{}, {}, {}, {}};
  for (int k = 0; k < FDIM; k += 4) {
    // B 4x16 fragment (layout mirrors A with M->N); loaded once, used 4x
    v2f b;
    b.x = W[(k + 2 * half) * FDIM + col0 + l16];
    b.y = W[(k + 2 * half + 1) * FDIM + col0 + l16];
#pragma unroll
    for (int rt = 0; rt < 4; ++rt) {
      // A 16x4: lanes 0-15 -> M=lane, VGPR0=K, VGPR1=K+1; lanes 16-31 K+2,K+3
      v2f a;
      a.x = As[rt * 16 + l16][k + 2 * half];
      a.y = As[rt * 16 + l16][k + 2 * half + 1];
      acc[rt] = __builtin_amdgcn_wmma_f32_16x16x4_f32(
          /*neg_a=*/false, a, /*neg_b=*/false, b,
          /*c_mod=*/(short)0, acc[rt], /*reuse_a=*/false, /*reuse_b=*/false);
    }
  }

  // C/D layout: VGPR r, lanes 0-15 -> M=r, N=lane; lanes 16-31 -> M=r+8
  // Fast path: whole 64-row tile in bounds (781/782 blocks) -> unguarded
  // stores, no per-element exec-mask churn.
  float* obase = out + (size_t)(row0 + 8 * half) * FDIM + col0 + l16;
  if (row0 + GROWS <= N) {
#pragma unroll
    for (int rt = 0; rt < 4; ++rt) {
#pragma unroll
      for (int r = 0; r < 8; ++r) {
        obase[(size_t)(rt * 16 + r) * FDIM] = acc[rt][r];
      }
    }
  } else {
#pragma unroll
    for (int rt = 0; rt < 4; ++rt) {
#pragma unroll
      for (int r = 0; r < 8; ++r) {
        int row = row0 + rt * 16 + r + 8 * half;
        if (row < N) out[(size_t)row * FDIM + col0 + l16] = acc[rt][r];
      }
    }
  }
}

// --------------------------- attention coefficients ------------------------
// Block = 128 threads = 4 waves; wave h reduces head h of one node.
__global__ void alpha_kernel(const float* __restrict__ xw,
                             const float* __restrict__ a_src,
                             const float* __restrict__ a_dst,
                             float* __restrict__ as_out,
                             float* __restrict__ ad_out, int N) {
  const int n = blockIdx.x;
  if (n >= N) return;
  const int h = threadIdx.x >> 5;
  const int c = threadIdx.x & 31;
  const float v = xw[(size_t)n * FDIM + h * CH + c];
  float s = v * a_src[h * CH + c];
  float d = v * a_dst[h * CH + c];
#pragma unroll
  for (int off = 16; off > 0; off >>= 1) {
    s += __shfl_down(s, off, 32);
    d += __shfl_down(d, off, 32);
  }
  if (c == 0) {
    as_out[n * HEADS + h] = s;
    ad_out[n * HEADS + h] = d;
  }
}

// ------------------------------- init --------------------------------------
__global__ void init_kernel(unsigned* __restrict__ mkey,
                            float* __restrict__ denom,
                            float* __restrict__ agg, int N) {
  const int i = blockIdx.x * blockDim.x + threadIdx.x;
  if (i < N * HEADS) {
    mkey[i] = FENC_NEG_INF;
    denom[i] = 0.0f;
  }
  if (i < N * FDIM) agg[i] = 0.0f;
}

// ---------------------- edge pass 1: leaky-relu + seg max ------------------
__global__ void edge_max_kernel(const float* __restrict__ as_,
                                const float* __restrict__ ad_,
                                const int* __restrict__ src,
                                const int* __restrict__ dst, int E, int N,
                                unsigned* __restrict__ mkey,
                                float4* __restrict__ estore) {
  const int i = blockIdx.x * blockDim.x + threadIdx.x;
  const int EP = E + N;
  if (i >= EP) return;
  const int s = (i < E) ? src[i] : (i - E);
  const int d = (i < E) ? dst[i] : (i - E);
  const float4 a = *(const float4*)(as_ + s * HEADS);
  const float4 b = *(const float4*)(ad_ + d * HEADS);
  float e0 = a.x + b.x, e1 = a.y + b.y, e2 = a.z + b.z, e3 = a.w + b.w;
  e0 = (e0 > 0.0f) ? e0 : NEG_SLOPE * e0;
  e1 = (e1 > 0.0f) ? e1 : NEG_SLOPE * e1;
  e2 = (e2 > 0.0f) ? e2 : NEG_SLOPE * e2;
  e3 = (e3 > 0.0f) ? e3 : NEG_SLOPE * e3;
  atomicMax(&mkey[d * HEADS + 0], fenc(e0));
  atomicMax(&mkey[d * HEADS + 1], fenc(e1));
  atomicMax(&mkey[d * HEADS + 2], fenc(e2));
  atomicMax(&mkey[d * HEADS + 3], fenc(e3));
  estore[i] = make_float4(e0, e1, e2, e3);
}

// ---------------------- edge pass 2: exp + seg sum -------------------------
__global__ void edge_expsum_kernel(float4* __restrict__ estore,
                                   const unsigned* __restrict__ mkey,
                                   const int* __restrict__ dst, int E, int N,
                                   float* __restrict__ denom) {
  const int i = blockIdx.x * blockDim.x + threadIdx.x;
  const int EP = E + N;
  if (i >= EP) return;
  const int d = (i < E) ? dst[i] : (i - E);
  const float4 e = estore[i];
  const float m0 = fdec(mkey[d * HEADS + 0]);
  const float m1 = fdec(mkey[d * HEADS + 1]);
  const float m2 = fdec(mkey[d * HEADS + 2]);
  const float m3 = fdec(mkey[d * HEADS + 3]);
  const float x0 = __expf(e.x - m0);
  const float x1 = __expf(e.y - m1);
  const float x2 = __expf(e.z - m2);
  const float x3 = __expf(e.w - m3);
  atomicAdd(&denom[d * HEADS + 0], x0);
  atomicAdd(&denom[d * HEADS + 1], x1);
  atomicAdd(&denom[d * HEADS + 2], x2);
  atomicAdd(&denom[d * HEADS + 3], x3);
  estore[i] = make_float4(x0, x1, x2, x3);
}

// ---------------------- edge pass 3: weighted aggregation ------------------
// One wave32 per edge. Lane c handles channel c of each of the 4 heads:
// gathers the full 512B row of xw[src] coalesced, 4 f32 atomics per lane.
__global__ void edge_agg_kernel(const float* __restrict__ xw,
                                const float4* __restrict__ ex,
                                const float* __restrict__ denom,
                                const int* __restrict__ src,
                                const int* __restrict__ dst, int E, int N,
                                float* __restrict__ agg) {
  const int gid = blockIdx.x * blockDim.x + threadIdx.x;
  const int wid = gid >> 5;
  const int lane = gid & 31;
  const int EP = E + N;
  if (wid >= EP) return;
  const int s = (wid < E) ? src[wid] : (wid - E);
  const int d = (wid < E) ? dst[wid] : (wid - E);
  const float4 e = ex[wid];
  const float4 dn = *(const float4*)(denom + d * HEADS);
  const float c0 = e.x / dn.x;
  const float c1 = e.y / dn.y;
  const float c2 = e.z / dn.z;
  const float c3 = e.w / dn.w;
  const float* xs = xw + (size_t)s * FDIM;
  float* od = agg + (size_t)d * FDIM;
  atomicAdd(od + 0 * CH + lane, xs[0 * CH + lane] * c0);
  atomicAdd(od + 1 * CH + lane, xs[1 * CH + lane] * c1);
  atomicAdd(od + 2 * CH + lane, xs[2 * CH + lane] * c2);
  atomicAdd(od + 3 * CH + lane, xs[3 * CH + lane] * c3);
}

// ------------------------------ bias + ELU ---------------------------------
__global__ void bias_elu_kernel(float* __restrict__ h,
                                const float* __restrict__ bias, int total) {
  const int i = blockIdx.x * blockDim.x + threadIdx.x;
  if (i >= total) return;
  const float v = h[i] + bias[i & (FDIM - 1)];
  h[i] = (v > 0.0f) ? v : (__expf(v) - 1.0f);
}

// -------------------- final linear + log_softmax ---------------------------
// One block (64 threads) per node; 40 logits, shared-mem reductions.
__global__ void linear_lsm_kernel(const float* __restrict__ h,
                                  const float* __restrict__ Wl,
                                  const float* __restrict__ bl,
                                  float* __restrict__ out, int N) {
  __shared__ float logits[NCLS];
  __shared__ float red[64];
  const int n = blockIdx.x;
  if (n >= N) return;
  const int t = threadIdx.x;
  const float* hr = h + (size_t)n * FDIM;
  if (t < NCLS) {
    float acc = bl[t];
#pragma unroll 8
    for (int k = 0; k < FDIM; ++k) acc += hr[k] * Wl[k * NCLS + t];
    logits[t] = acc;
  }
  __syncthreads();
  float v = (t < NCLS) ? logits[t] : -3.4e38f;
  red[t] = v;
  __syncthreads();
  for (int off = 32; off > 0; off >>= 1) {
    if (t < off) red[t] = fmaxf(red[t], red[t + off]);
    __syncthreads();
  }
  const float m = red[0];
  __syncthreads();
  red[t] = (t < NCLS) ? __expf(logits[t] - m) : 0.0f;
  __syncthreads();
  for (int off = 32; off > 0; off >>= 1) {
    if (t < off) red[t] += red[t + off];
    __syncthreads();
  }
  const float lse = m + __logf(red[0]);
  if (t < NCLS) out[(size_t)n * NCLS + t] = logits[t] - lse;
}

// ---------------------------------------------------------------------------
extern "C" void kernel_launch(void* const* d_in, const int* in_sizes, int n_in,
                              void* d_out, int out_size, void* d_ws,
                              size_t ws_size, hipStream_t stream) {
  const float* x = (const float*)d_in[0];
  const int* esrc = (const int*)d_in[1];
  const int* edst = (const int*)d_in[2];
  const int N = in_sizes[0] / FDIM;
  const int E = in_sizes[1];
  const int EP = E + N;

  // ---- carve workspace ----
  char* ws = (char*)d_ws;
  size_t off = 0;
  auto carve = [&](size_t bytes) -> void* {
    void* p = (void*)(ws + off);
    off = (off + bytes + 255) & ~(size_t)255;
    return p;
  };
  float* xw = (float*)carve((size_t)N * FDIM * sizeof(float));
  float* hA = (float*)carve((size_t)N * FDIM * sizeof(float));
  float* hB = (float*)carve((size_t)N * FDIM * sizeof(float));
  float* as_ = (float*)carve((size_t)N * HEADS * sizeof(float));
  float* ad_ = (float*)carve((size_t)N * HEADS * sizeof(float));
  unsigned* mkey = (unsigned*)carve((size_t)N * HEADS * sizeof(unsigned));
  float* denom = (float*)carve((size_t)N * HEADS * sizeof(float));
  float4* ex = (float4*)carve((size_t)EP * sizeof(float4));
  (void)ws_size;

  const dim3 B256(256);
  const unsigned gRows = (N + GROWS - 1) / GROWS;
  const unsigned gNodeF = (N * FDIM + 255) / 256;
  const unsigned gEdge = (EP + 255) / 256;
  const unsigned gAgg = (EP + 7) / 8;  // 8 waves per 256-thread block

  const float* hin = x;
  float* houts[3] = {hA, hB, hA};
  for (int l = 0; l < 3; ++l) {
    const float* W = (const float*)d_in[3 + l * 4];
    const float* av_s = (const float*)d_in[4 + l * 4];
    const float* av_d = (const float*)d_in[5 + l * 4];
    const float* bv = (const float*)d_in[6 + l * 4];
    float* hout = houts[l];

    gemm_xw_kernel<<<gRows, B256, 0, stream>>>(hin, W, xw, N);
    alpha_kernel<<<(unsigned)N, 128, 0, stream>>>(xw, av_s, av_d, as_, ad_, N);
    init_kernel<<<gNodeF, B256, 0, stream>>>(mkey, denom, hout, N);
    edge_max_kernel<<<gEdge, B256, 0, stream>>>(as_, ad_, esrc, edst, E, N,
                                                mkey, ex);
    edge_expsum_kernel<<<gEdge, B256, 0, stream>>>(ex, mkey, edst, E, N,
                                                   denom);
    edge_agg_kernel<<<gAgg, B256, 0, stream>>>(xw, ex, denom, esrc, edst, E,
                                               N, hout);
    bias_elu_kernel<<<gNodeF, B256, 0, stream>>>(hout, bv, N * FDIM);
    hin = hout;
  }

  linear_lsm_kernel<<<(unsigned)N, 64, 0, stream>>>(
      hin, (const float*)d_in[15], (const float*)d_in[16], (float*)d_out, N);
}